// TextureMartingaleModule_2199023256186
// MI455X (gfx1250) — compile-verified
//
#include <hip/hip_runtime.h>
#include <hip/hip_bf16.h>
#include <math.h>

// CDNA5 / gfx1250: wave32, WMMA 16x16x4 f32 used for dilated vertical box-sums.

typedef __attribute__((ext_vector_type(2))) float v2f;
typedef __attribute__((ext_vector_type(8))) float v8f;

#define EPS_F 1e-6f

template <int D>
__global__ __launch_bounds__(128)
void texfeat_kernel(const float* __restrict__ xin_all, float* __restrict__ out)
{
    constexpr int H   = 256, W = 256;
    constexpr int K1  = 2 * D + 1;          // kernel size per axis
    constexpr int PAD = D * D;              // conv padding = d^2
    constexpr int KK  = K1 * K1;            // taps per patch
    constexpr int TH  = 32, TW = 32;        // output tile per block
    constexpr int PH  = TH + 2 * PAD;       // padded tile rows
    constexpr int PW  = TW + 2 * PAD;       // padded tile cols
    constexpr int PWP = PW + 1;             // LDS row stride (bank stagger)
    constexpr int RP  = ((16 + 2 * PAD + 3) / 4) * 4;  // WMMA K-extent, mult of 4
    constexpr int NCH = RP / 4;             // number of 16x16x4 chunks
    constexpr int PHH = 16 + RP;            // H-sum rows incl. zero pad (>= PH)
    constexpr int HSP = TW + 1;             // H-sum row stride

    __shared__ float Tx[PH][PWP];           // padded x tile
    __shared__ float Tl[PH][PWP];           // x * log(x + eps)
    __shared__ float Hs[3][PHH][HSP];       // horiz dilated sums of x, x^2, x*logx

    const int tid = threadIdx.x;
    const int bc  = blockIdx.z;             // b*4 + c  (B=2, C=4)
    const int tx0 = blockIdx.x * TW;
    const int ty0 = blockIdx.y * TH;
    const float* __restrict__ xin = xin_all + (size_t)bc * (H * W);

    // ---- Stage A: load zero-padded tile, precompute x*log(x+eps) once ----
    for (int i = tid; i < PH * PW; i += 128) {
        const int r  = i / PW, c = i % PW;
        const int gy = ty0 - PAD + r;
        const int gx = tx0 - PAD + c;
        float v = 0.f;
        if (gy >= 0 && gy < H && gx >= 0 && gx < W) v = xin[gy * W + gx];
        Tx[r][c] = v;
        Tl[r][c] = v * __logf(v + EPS_F);
    }
    __syncthreads();

    // ---- Stage B: horizontal dilated box sums (separable pass 1) ----
    for (int i = tid; i < PHH * TW; i += 128) {
        const int r = i / TW, c = i % TW;
        float h1 = 0.f, h2 = 0.f, h3 = 0.f;
        if (r < PH) {
            #pragma unroll
            for (int j = 0; j < K1; ++j) {
                const float v = Tx[r][c + j * D];
                h1 += v;
                h2 = __builtin_fmaf(v, v, h2);
                h3 += Tl[r][c + j * D];
            }
        }
        Hs[0][r][c] = h1;   // zero rows >= PH keep 0-weight WMMA lanes NaN-free
        Hs[1][r][c] = h2;
        Hs[2][r][c] = h3;
    }
    __syncthreads();

    // ---- Stage C: vertical dilated box sums on the matrix pipe ----
    // S(16x16) = W(16xRP) x Hs(RPx16), W banded: W[m][r'] = 1 iff
    // (r'-m) in {0, D, 2D, ..., 2*PAD}. Chunked over K=4 WMMA ops.
    const int wave = tid >> 5, lane = tid & 31;
    const int sh = (wave >> 1) * 16;        // sub-tile row origin (0/16)
    const int sw = (wave & 1) * 16;         // sub-tile col origin (0/16)
    const int hi = lane >> 4;               // half-wave group
    const int n  = lane & 15;               // A's M index / B,C's N index

    // A fragment layout (32-bit A 16x4): lanes 0-15 M=lane, 16-31 M=lane-16;
    // VGPR0 holds K=0|2, VGPR1 holds K=1|3 (low|high half-wave).
    v2f afrag[NCH];
    #pragma unroll
    for (int ch = 0; ch < NCH; ++ch) {
        const int r0 = ch * 4 + 2 * hi;
        const int t0 = r0 - n;
        const int t1 = r0 + 1 - n;
        afrag[ch].x = (t0 >= 0 && t0 <= 2 * PAD && (t0 % D) == 0) ? 1.f : 0.f;
        afrag[ch].y = (t1 >= 0 && t1 <= 2 * PAD && (t1 % D) == 0) ? 1.f : 0.f;
    }

    v8f acc[3];
    #pragma unroll
    for (int f = 0; f < 3; ++f) {
        v8f s = {0.f, 0.f, 0.f, 0.f, 0.f, 0.f, 0.f, 0.f};
        #pragma unroll
        for (int ch = 0; ch < NCH; ++ch) {
            const int br = sh + ch * 4 + 2 * hi;
            v2f b;
            b.x = Hs[f][br][sw + n];
            b.y = Hs[f][br + 1][sw + n];
            // D = A*B + C : v_wmma_f32_16x16x4_f32
            s = __builtin_amdgcn_wmma_f32_16x16x4_f32(
                    false, afrag[ch], false, b, (short)0, s, false, false);
        }
        acc[f] = s;
    }

    // ---- Stage D: statistics + abs-deviation pass + martingale transform ----
    constexpr float KKf  = (float)KK;
    const float invK = 1.f / KKf;
    const float corr = KKf / (KKf - 1.f);       // unbiased std factor
    const float EH   = 0.6065306597126334f;     // exp(-0.5), theta = 1
    const size_t HW  = (size_t)H * W;
    float* __restrict__ o0 = out + (size_t)(bc * 4 + 0) * HW;  // contrast
    float* __restrict__ o1 = out + (size_t)(bc * 4 + 1) * HW;  // energy
    float* __restrict__ o2 = out + (size_t)(bc * 4 + 2) * HW;  // entropy
    float* __restrict__ o3 = out + (size_t)(bc * 4 + 3) * HW;  // homogeneity

    #pragma unroll
    for (int v = 0; v < 8; ++v) {
        const int m  = hi * 8 + v;              // C/D layout: M = v (+8 hi half)
        const int y  = sh + m;                  // tile-local row
        const int xc = sw + n;                  // tile-local col
        const float s1 = acc[0][v];
        const float s2 = acc[1][v];
        const float s3 = acc[2][v];

        const float mu = s1 * invK;
        float var_b = s2 * invK - mu * mu;
        var_b = var_b > 0.f ? var_b : 0.f;
        const float sd       = __builtin_sqrtf(var_b * corr) + EPS_F;
        const float contrast = var_b / (sd * sd);
        const float energy   = s2 * invK;
        const float entropy  = -s3 * invK;

        float ad = 0.f;                          // sum |p - mu| (not separable)
        for (int jy = 0; jy < K1; ++jy) {
            const float* rp = &Tx[y + jy * D][xc];
            #pragma unroll
            for (int jx = 0; jx < K1; ++jx)
                ad += __builtin_fabsf(rp[jx * D] - mu);
        }
        const float homog = 1.f / (1.f + ad * invK);

        const size_t pix = (size_t)(ty0 + y) * W + (tx0 + xc);
        o0[pix] = (contrast + EPS_F) * EH;
        o1[pix] = (energy   + EPS_F) * EH;
        o2[pix] = (entropy  + EPS_F) * EH;
        o3[pix] = (homog    + EPS_F) * EH;
    }
}

extern "C" void kernel_launch(void* const* d_in, const int* in_sizes, int n_in,
                              void* d_out, int out_size, void* d_ws, size_t ws_size,
                              hipStream_t stream) {
    (void)in_sizes; (void)n_in; (void)d_ws; (void)ws_size; (void)out_size;
    const float* x = (const float*)d_in[0];
    float* out = (float*)d_out;

    constexpr size_t PLANE = (size_t)2 * 16 * 256 * 256;  // one dilation's output
    dim3 grid(256 / 32, 256 / 32, 8);   // tilesX, tilesY, B*C
    dim3 blk(128);                      // 4 waves -> four 16x16 WMMA sub-tiles

    texfeat_kernel<1><<<grid, blk, 0, stream>>>(x, out + 0 * PLANE);
    texfeat_kernel<2><<<grid, blk, 0, stream>>>(x, out + 1 * PLANE);
    texfeat_kernel<3><<<grid, blk, 0, stream>>>(x, out + 2 * PLANE);
    texfeat_kernel<4><<<grid, blk, 0, stream>>>(x, out + 3 * PLANE);
}